// AffineTransformer_6442450944616
// MI455X (gfx1250) — compile-verified
//
#include <hip/hip_runtime.h>

// Problem constants (match reference)
#define SRC  64
#define PAD  32
#define WIN  128   // SRC + 2*PAD
#define HIN  128
#define TW   128   // SRC*2
#define TH   128

typedef float v4f __attribute__((ext_vector_type(4)));

// LDS byte-offset of a shared-memory pointer (generic -> addrspace(3) -> u32)
__device__ __forceinline__ unsigned lds_off(const void* p) {
    return (unsigned)(unsigned long long)(const __attribute__((address_space(3))) char*)p;
}

// One bilinear tap: returns {fill, stroke} via a single ds_load_b64 from the
// interleaved tile, or {0,0} outside the (unpadded) source region.
__device__ __forceinline__ float2 fetch_tap(const float2* __restrict__ s, int x, int y) {
    const unsigned ux = (unsigned)(x - PAD);
    const unsigned uy = (unsigned)(y - PAD);
    float2 v = make_float2(0.0f, 0.0f);
    if (ux < (unsigned)SRC && uy < (unsigned)SRC) v = s[uy * SRC + ux];
    return v;
}

__global__ __launch_bounds__(256) void affine_sampler_kernel(
    const float* __restrict__ aff,     // (N,6)
    const float* __restrict__ fill,    // (N,64,64)
    const float* __restrict__ stroke,  // (N,64,64)
    float* __restrict__ out,           // fill_out (N,128,128) ++ stroke_out (N,128,128)
    int nimg)
{
    // Interleaved tiles: sT[e] = { fill[e], stroke[e] }  (32 KB)
    __shared__ float2 sT[SRC * SRC];

    const int n   = blockIdx.x;
    const int tid = threadIdx.x;

    // ---- Stage both 64x64 tiles into LDS, interleaved, via CDNA5 async DMA.
    // Per-lane LDS destination addressing lets the DMA do the interleave:
    //   fill element e   -> LDS byte e*8
    //   stroke element e -> LDS byte e*8 + 4
    const float* gF = fill   + (size_t)n * (SRC * SRC);
    const float* gS = stroke + (size_t)n * (SRC * SRC);
    const unsigned lT = lds_off(sT);
#pragma unroll
    for (int i = 0; i < 16; ++i) {
        const unsigned e    = (unsigned)(i * 256 + tid);
        const unsigned goff = e * 4u;
        asm volatile("global_load_async_to_lds_b32 %0, %1, %2"
                     :: "v"(lT + e * 8u),      "v"(goff), "s"(gF) : "memory");
        asm volatile("global_load_async_to_lds_b32 %0, %1, %2"
                     :: "v"(lT + e * 8u + 4u), "v"(goff), "s"(gS) : "memory");
    }

    // ---- Affine coefficients (uniform across block -> scalar loads) ----
    const float o0 = aff[n * 6 + 0];
    const float o1 = aff[n * 6 + 1];
    const float o2 = aff[n * 6 + 2];
    const float o3 = aff[n * 6 + 3];
    const float o4 = aff[n * 6 + 4];
    const float o5 = aff[n * 6 + 5];
    const float a00 = 2.0f / (1.0f + expf(-o0));   // 2*sigmoid
    const float a11 = 2.0f / (1.0f + expf(-o1));
    const float a01 = 2.0f * tanhf(o2);
    const float a10 = 2.0f * tanhf(o3);
    const float a02 = tanhf(o4);
    const float a12 = tanhf(o5);

    // Wait for DMA completion, then make LDS visible to all waves
#if __has_builtin(__builtin_amdgcn_s_wait_asynccnt)
    __builtin_amdgcn_s_wait_asynccnt(0);
#else
    asm volatile("s_wait_asynccnt 0" ::: "memory");
#endif
    __syncthreads();

    const size_t fill_base   = (size_t)n * (TH * TW);
    const size_t stroke_base = (size_t)nimg * (TH * TW) + fill_base;

    // ---- 128x128 output, 4 consecutive pixels per thread per iteration ----
#pragma unroll 2
    for (int k = 0; k < 16; ++k) {
        const int p4  = k * 1024 + tid * 4;        // first pixel of this group of 4
        const int ty  = p4 >> 7;
        const int tx0 = p4 & 127;

        const float Y = ((float)ty + 0.5f) * (2.0f / TH) - 1.0f;

        v4f fo, so;
#pragma unroll
        for (int j = 0; j < 4; ++j) {
            const float X  = ((float)(tx0 + j) + 0.5f) * (2.0f / TW) - 1.0f;
            const float gx = a00 * X + a01 * Y + a02;
            const float gy = a10 * X + a11 * Y + a12;
            const float ix = ((gx + 1.0f) * (float)WIN - 1.0f) * 0.5f;
            const float iy = ((gy + 1.0f) * (float)HIN - 1.0f) * 0.5f;

            const float x0f = floorf(ix);
            const float y0f = floorf(iy);
            const float wx  = ix - x0f;
            const float wy  = iy - y0f;
            const int   x0  = (int)x0f;
            const int   y0  = (int)y0f;

            const float w00 = (1.0f - wx) * (1.0f - wy);
            const float w10 = wx * (1.0f - wy);
            const float w01 = (1.0f - wx) * wy;
            const float w11 = wx * wy;

            const float2 v00 = fetch_tap(sT, x0,     y0);
            const float2 v10 = fetch_tap(sT, x0 + 1, y0);
            const float2 v01 = fetch_tap(sT, x0,     y0 + 1);
            const float2 v11 = fetch_tap(sT, x0 + 1, y0 + 1);

            fo[j] = w00 * v00.x + w10 * v10.x + w01 * v01.x + w11 * v11.x;
            so[j] = w00 * v00.y + w10 * v10.y + w01 * v01.y + w11 * v11.y;
        }

        // Coalesced 16B non-temporal stores (output >> L2; keep L2 for input tiles)
        __builtin_nontemporal_store(fo, (v4f*)(out + fill_base   + p4));
        __builtin_nontemporal_store(so, (v4f*)(out + stroke_base + p4));
    }
}

extern "C" void kernel_launch(void* const* d_in, const int* in_sizes, int n_in,
                              void* d_out, int out_size, void* d_ws, size_t ws_size,
                              hipStream_t stream) {
    const float* aff    = (const float*)d_in[0];  // (N,6)
    const float* fill   = (const float*)d_in[1];  // (N,64,64)
    const float* stroke = (const float*)d_in[2];  // (N,64,64)
    // d_in[3] = targetsize (unused; always SRC)
    const int nimg = in_sizes[0] / 6;

    affine_sampler_kernel<<<dim3(nimg), dim3(256), 0, stream>>>(
        aff, fill, stroke, (float*)d_out, nimg);
}